// HadamardMultiplier_481036337409
// MI455X (gfx1250) — compile-verified
//
#include <hip/hip_runtime.h>

// MI455X (gfx1250) Hadamard transform:
//   y[t, k*1024 + m] = (1/sqrt(12288)) * sum_l had_K[k,l] * sum_s (-1)^popc(m&s) x[t, l*1024+s]
// Memory-bound (0.8 GB @ 23.3 TB/s ~ 35us). FWHT butterflies on VALU (cheap,
// O(n log n)); the 12x12 residual matmul (majority of FLOPs) on WMMA f32 16x16x4.

typedef float v2f __attribute__((ext_vector_type(2)));
typedef float v8f __attribute__((ext_vector_type(8)));

#define T_ROWS  8192
#define N_COLS  12288
#define KHAD    12
#define CHUNK   1024
#define NCHUNK  12
#define ZPITCH  36     // row pitch (floats) of 32x32 transpose scratch; 36*4B = 16B aligned
#define CPITCH  1168   // per-chunk LDS region (floats): >= 32*36, %64 == 16 (bank stagger)

__device__ __forceinline__ void fwht32(float v[32]) {
  #pragma unroll
  for (int bit = 1; bit < 32; bit <<= 1) {
    #pragma unroll
    for (int i = 0; i < 32; ++i) {
      if ((i & bit) == 0) {
        float a = v[i];
        float b = v[i | bit];
        v[i]       = a + b;
        v[i | bit] = a - b;
      }
    }
  }
}

// One 16-column output tile: D(16x16) = A(had_K/sqrt(n), 16x12 padded) x B(12x16 from LDS).
__device__ __forceinline__ void do_tile(const float* __restrict__ smem,
                                        const v2f A[3],
                                        float* __restrict__ out,
                                        size_t outBase, int tile,
                                        int mrow, int half) {
  const int m0 = tile * 16;
  v8f c = {0.f, 0.f, 0.f, 0.f, 0.f, 0.f, 0.f, 0.f};
  #pragma unroll
  for (int kb = 0; kb < 3; ++kb) {
    // B: 4x16 f32 (2 VGPRs): reg q, lanes 0-15 -> K row, lanes 16-31 -> K row +2
    v2f b;
    #pragma unroll
    for (int q = 0; q < 2; ++q) {
      const int krow = kb * 4 + half * 2 + q;          // chunk index l (0..11)
      b[q] = smem[krow * CPITCH + m0 + mrow];
    }
    c = __builtin_amdgcn_wmma_f32_16x16x4_f32(
        /*neg_a=*/false, A[kb], /*neg_b=*/false, b,
        /*c_mod=*/(short)0, c, /*reuse_a=*/false, /*reuse_b=*/false);
  }

  // D 16x16: VGPR j -> lanes 0-15 row M=j, lanes 16-31 row M=j+8.
  // Regs 0-3 are rows 0-3 / 8-11 (always valid); regs 4-7 are rows 4-7 for
  // lanes 0-15 but rows 12-15 (nonexistent) for lanes 16-31 -> one exec toggle.
  const size_t o = outBase + (size_t)m0 + (size_t)mrow;
  #pragma unroll
  for (int j = 0; j < 4; ++j)
    out[o + (size_t)(j + half * 8) * CHUNK] = c[j];
  if (half == 0) {
    #pragma unroll
    for (int j = 4; j < 8; ++j)
      out[o + (size_t)j * CHUNK] = c[j];
  }
}

__global__ __launch_bounds__(384) void hadamard_wmma_kernel(
    const float* __restrict__ x,
    const float* __restrict__ hadK,
    float* __restrict__ out) {
  __shared__ __align__(16) float smem[NCHUNK * CPITCH];  // 56,064 B

  const int t    = blockIdx.x;
  const int tid  = threadIdx.x;
  // Wave id is uniform across the wave: pin it to an SGPR so all wave-level
  // control flow (tile ownership, tail) is scalar, not exec-masked.
  const int wave = __builtin_amdgcn_readfirstlane(tid >> 5);  // 0..11
  const int lane = tid & 31;

  // ---------------- Phase 1: FWHT-1024 on chunk l = wave ----------------
  // Chunk viewed as 32x32: M[r][c] = x[base + r*32 + c]; FWHT along c (regs),
  // LDS transpose, FWHT along r (regs). Output index m = mr*32 + mc.
  {
    const int l = wave;
    const float4* src = reinterpret_cast<const float4*>(
        x + (size_t)t * N_COLS + (size_t)l * CHUNK + (size_t)lane * 32);
    float v[32];
    #pragma unroll
    for (int j = 0; j < 8; ++j) {        // coalesced global_load_b128
      float4 q = src[j];
      v[4*j+0] = q.x; v[4*j+1] = q.y; v[4*j+2] = q.z; v[4*j+3] = q.w;
    }

    fwht32(v);                           // FWHT over c (low 5 bits of s)

    float* zbase = smem + l * CPITCH;
    #pragma unroll
    for (int j = 0; j < 8; ++j) {        // Z row-major, 16B-aligned b128 stores
      float4 q;
      q.x = v[4*j+0]; q.y = v[4*j+1]; q.z = v[4*j+2]; q.w = v[4*j+3];
      *reinterpret_cast<float4*>(zbase + lane * ZPITCH + 4*j) = q;
    }

    float w2[32];                        // transpose read: lane -> column mc
    #pragma unroll
    for (int r = 0; r < 32; ++r)
      w2[r] = zbase[r * ZPITCH + lane];  // lanes consecutive -> conflict-free

    fwht32(w2);                          // FWHT over r (high 5 bits of s)

    #pragma unroll
    for (int mr = 0; mr < 32; ++mr)      // final[l][m], m = mr*32 + lane
      zbase[mr * 32 + lane] = w2[mr];
  }
  __syncthreads();

  // ---------------- Phase 2: D = (had_K/sqrt(n)) x final via WMMA ----------------
  // A: 16x4 f32 per WMMA (2 VGPRs): lanes 0-15 hold M=lane, regs {K0,K1};
  //    lanes 16-31 hold M=lane-16, regs {K2,K3}. Rows/cols >= 12 zero-padded.
  const float scale = 0.009021097956087902f;  // 1/sqrt(12288)
  const int mrow = lane & 15;
  const int half = lane >> 4;

  v2f A[3];
  #pragma unroll
  for (int kb = 0; kb < 3; ++kb) {
    #pragma unroll
    for (int q = 0; q < 2; ++q) {
      const int col = kb * 4 + half * 2 + q;
      A[kb][q] = (mrow < KHAD && col < KHAD) ? hadK[mrow * KHAD + col] * scale : 0.0f;
    }
  }

  const size_t outBase = (size_t)t * N_COLS;
  // 64 tiles over 12 waves: 5 full rounds for everyone + 1 tail tile for waves 0-3.
  #pragma unroll
  for (int i = 0; i < 5; ++i)
    do_tile(smem, A, out, outBase, wave + NCHUNK * i, mrow, half);
  if (wave < 4)   // scalar-uniform branch (wave is SGPR)
    do_tile(smem, A, out, outBase, 60 + wave, mrow, half);
}

extern "C" void kernel_launch(void* const* d_in, const int* in_sizes, int n_in,
                              void* d_out, int out_size, void* d_ws, size_t ws_size,
                              hipStream_t stream) {
  (void)in_sizes; (void)n_in; (void)d_ws; (void)ws_size; (void)out_size;
  const float* x    = (const float*)d_in[0];   // (8192, 12288) f32
  const float* hadK = (const float*)d_in[1];   // (12, 12) f32 (+/-1)
  float* out        = (float*)d_out;           // (8192, 12288) f32

  dim3 grid(T_ROWS);
  dim3 block(384);   // 12 wave32s: one wave per 1024-chunk
  hipLaunchKernelGGL(hadamard_wmma_kernel, grid, block, 0, stream, x, hadK, out);
}